// CrossScan_90898687852701
// MI455X (gfx1250) — compile-verified
//
#include <hip/hip_runtime.h>

// CrossScan for x: (B=8, C=192, H=128, W=128) f32 -> (4, B, H*W, C) f32.
// Pure permutation: bandwidth-bound (96 MiB in, 384 MiB out ~ 21.6 us at 23.3 TB/s).
// Strategy: per block, async-DMA a 32-channel x 128-w tile (16 KB) global->LDS
// (CDNA5 ASYNCcnt path), transpose via LDS reads, fan each value out to all 4
// scans as coalesced 128B-per-8-lane nontemporal b128 stores.

typedef float v4f __attribute__((ext_vector_type(4)));
typedef int   v4i __attribute__((ext_vector_type(4)));

#define BATCH  8
#define CHAN   192
#define HH     128
#define WW     128
#define HW     (HH * WW)            // 16384
#define C_TILE 32
#define NCB    (CHAN / C_TILE)      // 6 channel-blocks
#define PITCH  132                  // LDS row pitch in floats: 528B = 33*16B (16B aligned), 132%64=4 (<=2-way conflicts)
#define SCAN_ELEMS ((size_t)BATCH * HW * CHAN)   // 25,165,824 per scan

#if defined(__HIP_DEVICE_COMPILE__) && __has_builtin(__builtin_amdgcn_global_load_async_to_lds_b128)
#define USE_ASYNC_LDS 1
#else
#define USE_ASYNC_LDS 0
#endif

__global__ __launch_bounds__(256) void cross_scan_kernel(
    const float* __restrict__ x, float* __restrict__ out)
{
    __shared__ float lds[C_TILE * PITCH];   // 16,896 B

    const int tid = threadIdx.x;
    int bx = blockIdx.x;
    const int cb = bx % NCB;  bx /= NCB;    // channel block 0..5
    const int h  = bx % HH;   bx /= HH;     // image row 0..127
    const int b  = bx;                      // batch 0..7
    const int c0 = cb * C_TILE;

    // Source tile base: x[b][c0][h][0]
    const float* src = x + (((size_t)b * CHAN + c0) * HW + (size_t)h * WW);

    // ---- Stage 1: global -> LDS, layout lds[c][w] (w-contiguous rows) ----
#if USE_ASYNC_LDS
    {
#pragma unroll
        for (int i = 0; i < 4; ++i) {
            const int e  = tid + 256 * i;   // 0..1023 tile b128-chunks
            const int c  = e >> 5;          // 0..31 channel within tile
            const int wq = e & 31;          // 0..31 w-quad
            v4i* gp = (v4i*)(src + (size_t)c * HW + (size_t)wq * 4);
            v4i* lp = (v4i*)(&lds[c * PITCH + wq * 4]);
            __builtin_amdgcn_global_load_async_to_lds_b128(gp, lp,
                                                           /*offset=*/0,
                                                           /*cpol=*/0);
        }
#if __has_builtin(__builtin_amdgcn_s_wait_asynccnt)
        __builtin_amdgcn_s_wait_asynccnt(0);
#else
        asm volatile("s_wait_asynccnt 0" ::: "memory");
#endif
    }
#else
#pragma unroll
    for (int i = 0; i < 4; ++i) {
        const int e  = tid + 256 * i;
        const int c  = e >> 5;
        const int wq = e & 31;
        v4f v = __builtin_nontemporal_load(
            (const v4f*)(src + (size_t)c * HW + (size_t)wq * 4));
        *(v4f*)&lds[c * PITCH + wq * 4] = v;
    }
#endif
    __syncthreads();

    // ---- Stage 2: LDS transpose reads, 4-way fan-out nontemporal stores ----
    float* o1 = out;                    // scan1: l = h*W + w
    float* o2 = out + SCAN_ELEMS;       // scan2: l = h*W + (W-1-w)
    float* o3 = out + 2 * SCAN_ELEMS;   // scan3: l = w*H + h
    float* o4 = out + 3 * SCAN_ELEMS;   // scan4: l = w*H + (H-1-h)
    const size_t lb = (size_t)b * HW;

#pragma unroll
    for (int i = 0; i < 4; ++i) {
        const int p = tid + 256 * i;    // 0..1023 (w, c-quad) pairs
        const int q = p & 7;            // c-quad within tile (8 lanes span 32 ch = 128B)
        const int w = p >> 3;           // 0..127

        v4f v;                          // transpose: gather 4 consecutive channels
        v.x = lds[(4 * q + 0) * PITCH + w];
        v.y = lds[(4 * q + 1) * PITCH + w];
        v.z = lds[(4 * q + 2) * PITCH + w];
        v.w = lds[(4 * q + 3) * PITCH + w];

        const int    cc = c0 + 4 * q;
        const size_t l1 = lb + (size_t)h * WW + w;
        const size_t l2 = lb + (size_t)h * WW + (WW - 1 - w);
        const size_t l3 = lb + (size_t)w * HH + h;
        const size_t l4 = lb + (size_t)w * HH + (HH - 1 - h);

        __builtin_nontemporal_store(v, (v4f*)(o1 + l1 * CHAN + cc));
        __builtin_nontemporal_store(v, (v4f*)(o2 + l2 * CHAN + cc));
        __builtin_nontemporal_store(v, (v4f*)(o3 + l3 * CHAN + cc));
        __builtin_nontemporal_store(v, (v4f*)(o4 + l4 * CHAN + cc));
    }
}

extern "C" void kernel_launch(void* const* d_in, const int* in_sizes, int n_in,
                              void* d_out, int out_size, void* d_ws, size_t ws_size,
                              hipStream_t stream) {
    const float* x = (const float*)d_in[0];
    float* out = (float*)d_out;
    const int blocks = BATCH * HH * NCB;   // 6144
    hipLaunchKernelGGL(cross_scan_kernel, dim3(blocks), dim3(256), 0, stream,
                       x, out);
}